// StrongBlock_44272522887311
// MI455X (gfx1250) — compile-verified
//
#include <hip/hip_runtime.h>
#include <hip/hip_bf16.h>

// ---------------------------------------------------------------------------
// 21-qubit strongly-entangling layers, fp32, WMMA f32 16x16x4 on CDNA5.
//   - gates fused 4-at-a-time into 16x16 complex unitaries (Kronecker)
//   - 2 LDS-resident sweeps per depth (hi bits / lo bits)
//   - all CNOTs applied as index permutations folded into stores
// ---------------------------------------------------------------------------

#define NQ    21
#define SDIM  (1 << NQ)          // 2^21 amplitudes
#define NB    2                  // batch
#define NDEP  3

typedef float v2f __attribute__((ext_vector_type(2)));
typedef float v8f __attribute__((ext_vector_type(8)));

__device__ __forceinline__ v8f wmma4(v2f a, v2f b, v8f c) {
  // D(16x16,f32) = A(16x4,f32) * B(4x16,f32) + C
  return __builtin_amdgcn_wmma_f32_16x16x4_f32(false, a, false, b, (short)0, c,
                                               false, false);
}

// ---------------------------------------------------------------------------
// K0: fused per-gate 2x2 unitary  U = Rx(c/2) Ry(b/2) Rz(a/2)  (half angles)
// layout: gates[g*8] = {r00,i00,r01,i01,r10,i10,r11,i11}, g = d*21+q
// ---------------------------------------------------------------------------
__global__ void k_gates(const float* __restrict__ params,
                        float* __restrict__ gates) {
  int id = blockIdx.x * blockDim.x + threadIdx.x;
  if (id >= NDEP * NQ) return;
  float a = params[id * 3 + 0] * 0.5f;
  float b = params[id * 3 + 1] * 0.5f;
  float c = params[id * 3 + 2] * 0.5f;
  float ca = cosf(a), sa = sinf(a);
  float cb = cosf(b), sb = sinf(b);
  float cc = cosf(c), sc = sinf(c);
  float* u = gates + id * 8;
  u[0] = cc * cb * ca - sc * sb * sa;      // r00
  u[1] = -(cc * cb * sa + sc * sb * ca);   // i00
  u[2] = -cc * sb * ca + sc * cb * sa;     // r01
  u[3] = -(cc * sb * sa + sc * cb * ca);   // i01
  u[4] = cc * sb * ca - sc * cb * sa;      // r10
  u[5] = -(cc * sb * sa + sc * cb * ca);   // i10
  u[6] = cc * cb * ca - sc * sb * sa;      // r11
  u[7] = cc * cb * sa + sc * sb * ca;      // i11
}

// ---------------------------------------------------------------------------
// K1: fused 16x16 group matrices (Kronecker of up to 4 gates, identity pad)
// groups per depth: 0..2 hi-pass, 3..5 lo-pass.
// layout: gmats[(d*6+g)*512 + plane*256 + r*16 + c], plane 0=re 1=im
// ---------------------------------------------------------------------------
__global__ void k_groups(const float* __restrict__ gates,
                         float* __restrict__ gmats) {
  int id = blockIdx.x * blockDim.x + threadIdx.x;
  if (id >= NDEP * 6 * 256) return;
  int d = id / 1536;
  int rem = id - d * 1536;
  int g = rem >> 8;
  int rc = rem & 255;
  int r = rc >> 4, c = rc & 15;
  // slot s=0 is the MSB of the 4-bit group index
  const int slot[6][4] = {
      {7, 8, 9, 10},    // hi k=2  (h-bits 3..0 = qubits 7..10)
      {3, 4, 5, 6},     // hi k=6  (h-bits 7..4 = qubits 3..6)
      {0, 1, 2, -1},    // hi k=9  (h-bits 10..7 = qubits 0..2, id on bit7)
      {17, 18, 19, 20}, // lo k=0
      {13, 14, 15, 16}, // lo k=4
      {11, 12, -1, -1}  // lo k=6  (bits 9..6 = q11,q12, id, id)
  };
  float pr = 1.0f, pi = 0.0f;
#pragma unroll
  for (int s = 0; s < 4; ++s) {
    int q = slot[g][s];
    int rb = (r >> (3 - s)) & 1;
    int cb = (c >> (3 - s)) & 1;
    float fr, fi;
    if (q < 0) {
      fr = (rb == cb) ? 1.0f : 0.0f;
      fi = 0.0f;
    } else {
      const float* u = gates + (d * NQ + q) * 8 + (rb * 2 + cb) * 2;
      fr = u[0];
      fi = u[1];
    }
    float nr = pr * fr - pi * fi;
    float ni = pr * fi + pi * fr;
    pr = nr;
    pi = ni;
  }
  float* m = gmats + (d * 6 + g) * 512;
  m[r * 16 + c] = pr;
  m[256 + r * 16 + c] = pi;
}

// ---------------------------------------------------------------------------
// Apply one fused 16x16 complex unitary to an LDS-resident block.
// Local linear space of L elements; the 4 gate bits sit at element-bits
// [kk+3..kk]. Complex matmul = 4 real WMMA products, K swept 4 at a time.
// All 256 threads (8 waves) must call this. The tile loop is a uniform
// counted loop (L is a multiple of 2048) so EXEC is structurally all-ones
// around every WMMA, as the ISA requires.
// ---------------------------------------------------------------------------
__device__ __forceinline__ void apply_group16(float* sre, float* sim,
                                              const float* mre,
                                              const float* mim, int L, int kk,
                                              int tid) {
  const int lane = tid & 31;
  const int wave = tid >> 5;
  const int n = lane & 15;     // A row / B,D column within tile
  const int half = lane >> 4;  // K half-select per the f32 VGPR layout
  v2f ar[4], ai[4], an[4];
#pragma unroll
  for (int K = 0; K < 4; ++K) {
    int j0 = 4 * K + half * 2;
    v2f t;
    t.x = mre[n * 16 + j0];
    t.y = mre[n * 16 + j0 + 1];
    ar[K] = t;
    v2f u;
    u.x = mim[n * 16 + j0];
    u.y = mim[n * 16 + j0 + 1];
    ai[K] = u;
    v2f v;
    v.x = -u.x;
    v.y = -u.y;
    an[K] = v;  // f32 WMMA has no A-negate modifier; negate in VALU
  }
  const int nt8 = L >> 11;  // ntiles / 8 waves (exact: L is 2048 or 8192)
  const int mask = (1 << kk) - 1;
  const int rs = 1 << kk;
  for (int it = 0; it < nt8; ++it) {
    int t = (it << 3) + wave;
    int cN = t * 16 + n;
    int colbase = ((cN >> kk) << (kk + 4)) + (cN & mask);
    v8f dr = {0.f, 0.f, 0.f, 0.f, 0.f, 0.f, 0.f, 0.f};
    v8f di = {0.f, 0.f, 0.f, 0.f, 0.f, 0.f, 0.f, 0.f};
#pragma unroll
    for (int K = 0; K < 4; ++K) {
      int r0 = 4 * K + half * 2;
      int a0 = colbase + r0 * rs;
      v2f br;
      br.x = sre[a0];
      br.y = sre[a0 + rs];
      v2f bi;
      bi.x = sim[a0];
      bi.y = sim[a0 + rs];
      dr = wmma4(ar[K], br, dr);  // +Mr*Xr
      dr = wmma4(an[K], bi, dr);  // -Mi*Xi
      di = wmma4(ar[K], bi, di);  // +Mr*Xi
      di = wmma4(ai[K], br, di);  // +Mi*Xr
    }
#pragma unroll
    for (int v = 0; v < 8; ++v) {
      int row = v + half * 8;
      int addr = colbase + row * rs;
      sre[addr] = dr[v];
      sim[addr] = di[v];
    }
  }
}

// ---------------------------------------------------------------------------
// PassHi: block = all 2048 hi values (bits 10..20) x 4 consecutive lo values.
// Applies hi gate groups, then CNOT chain (0,1)..(9,10) as a permutation
// folded into the store. In-place safe (read set == write set per block).
// ---------------------------------------------------------------------------
__global__ __launch_bounds__(256) void pass_hi(
    const float* __restrict__ src_re, const float* __restrict__ src_im,
    int sb, float* __restrict__ dst_re, float* __restrict__ dst_im, int db,
    const float* __restrict__ gmat) {
  __shared__ __align__(16) float sre[8192];
  __shared__ __align__(16) float sim[8192];
  __shared__ float mre[256];
  __shared__ float mim[256];
  const int tid = threadIdx.x;
  const int b = blockIdx.x >> 8;
  const int l0 = (blockIdx.x & 255) * 4;
  const float* pre = src_re + (long)b * sb;
  const float* pim = src_im + (long)b * sb;
  for (int h = tid; h < 2048; h += 256) {
    long g = ((long)h << 10) + l0;
    float4 r4 = *(const float4*)(pre + g);
    float4 i4 = *(const float4*)(pim + g);
    *(float4*)&sre[h * 4] = r4;
    *(float4*)&sim[h * 4] = i4;
  }
  for (int gi = 0; gi < 3; ++gi) {
    int kk = (gi == 0) ? 2 : (gi == 1) ? 6 : 9;  // h-bit k + 2 (l' bits)
    __syncthreads();
    const float* gm = gmat + gi * 512;
    mre[tid] = gm[tid];
    mim[tid] = gm[256 + tid];
    __syncthreads();
    apply_group16(sre, sim, mre, mim, 8192, kk, tid);
  }
  __syncthreads();
  float* qre = dst_re + (long)b * db;
  float* qim = dst_im + (long)b * db;
  for (int h = tid; h < 2048; h += 256) {
    // source index under CNOT chain (0,1)..(9,10); apply inverses in reverse
    int y = h;
#pragma unroll
    for (int q = 9; q >= 0; --q) y ^= ((y >> (10 - q)) & 1) << (9 - q);
    float4 r4 = *(const float4*)&sre[y * 4];
    float4 i4 = *(const float4*)&sim[y * 4];
    long g = ((long)h << 10) + l0;
    *(float4*)(qre + g) = r4;
    *(float4*)(qim + g) = i4;
  }
}

// ---------------------------------------------------------------------------
// PassLo: block = 2048 contiguous amplitudes (two 1024-element lo spaces).
// Applies lo gate groups, then CNOT(10,11) + chain (11,12)..(19,20) folded
// into the LDS gather, and ring CNOT(20,0) folded into a scattered store.
// Optionally writes the final interleaved (B, DIM, 2) output.
// ---------------------------------------------------------------------------
__global__ __launch_bounds__(256) void pass_lo(
    const float* __restrict__ src_re, const float* __restrict__ src_im,
    int sb, float* __restrict__ dst_re, float* __restrict__ dst_im, int db,
    float* __restrict__ out_inter, int interleaved,
    const float* __restrict__ gmat) {
  __shared__ __align__(16) float sre[2048];
  __shared__ __align__(16) float sim[2048];
  __shared__ float mre[256];
  __shared__ float mim[256];
  const int tid = threadIdx.x;
  const int b = blockIdx.x >> 10;
  const int chunk = blockIdx.x & 1023;
  const long base = (long)chunk * 2048;
  const float* pre = src_re + (long)b * sb + base;
  const float* pim = src_im + (long)b * sb + base;
  for (int x4 = tid; x4 < 512; x4 += 256) {
    float4 r4 = *(const float4*)(pre + x4 * 4);
    float4 i4 = *(const float4*)(pim + x4 * 4);
    *(float4*)&sre[x4 * 4] = r4;
    *(float4*)&sim[x4 * 4] = i4;
  }
  for (int gi = 0; gi < 3; ++gi) {
    int kk = (gi == 0) ? 0 : (gi == 1) ? 4 : 6;
    __syncthreads();
    const float* gm = gmat + gi * 512;
    mre[tid] = gm[tid];
    mim[tid] = gm[256 + tid];
    __syncthreads();
    apply_group16(sre, sim, mre, mim, 2048, kk, tid);
  }
  __syncthreads();
  for (int x = tid; x < 2048; x += 256) {
    int sub = x >> 10;  // global bit 10
    int lo = x & 1023;
    // source under chain (11,12)..(19,20): apply in reverse temporal order
    int y = lo;
#pragma unroll
    for (int q = 19; q >= 11; --q) y ^= ((y >> (20 - q)) & 1) << (19 - q);
    y ^= sub << 9;  // CNOT(10,11): flip bit 9 when bit 10 set (applied first)
    float vr = sre[(sub << 10) | y];
    float vi = sim[(sub << 10) | y];
    long gidx = base + x;
    long oidx = gidx ^ ((long)(x & 1) << 20);  // ring CNOT(20,0) scatter
    if (interleaved) {
      float2 o;
      o.x = vr;
      o.y = vi;
      *(float2*)(out_inter + ((long)b * SDIM + oidx) * 2) = o;
    } else {
      dst_re[(long)b * db + oidx] = vr;
      dst_im[(long)b * db + oidx] = vi;
    }
  }
}

// ---------------------------------------------------------------------------
// Host-side orchestration.
//   ws layout (floats): [0..503]   fused 2x2 gates
//                       [512..9727] fused 16x16 group matrices
//                       [16384..]  plane buffer W1: [b][re|im][SDIM]  (32 MiB)
//   d_out doubles as plane buffer W2 until the final interleaving pass.
// ---------------------------------------------------------------------------
extern "C" void kernel_launch(void* const* d_in, const int* in_sizes, int n_in,
                              void* d_out, int out_size, void* d_ws,
                              size_t ws_size, hipStream_t stream) {
  (void)in_sizes;
  (void)n_in;
  (void)out_size;
  (void)ws_size;
  const float* state_re = (const float*)d_in[0];
  const float* state_im = (const float*)d_in[1];
  const float* params = (const float*)d_in[2];
  float* out = (float*)d_out;
  float* ws = (float*)d_ws;

  float* gates = ws;
  float* gmats = ws + 512;
  float* w1 = ws + 16384;  // [b][2][SDIM]
  float* w2 = out;         // reused as planes until final pass

  k_gates<<<1, 64, 0, stream>>>(params, gates);
  k_groups<<<18, 256, 0, stream>>>(gates, gmats);

  const int PS = 2 * SDIM;  // plane batch stride
  // depth 0
  pass_hi<<<512, 256, 0, stream>>>(state_re, state_im, SDIM, w1, w1 + SDIM, PS,
                                   gmats + 0 * 6 * 512);
  pass_lo<<<2048, 256, 0, stream>>>(w1, w1 + SDIM, PS, w2, w2 + SDIM, PS,
                                    nullptr, 0, gmats + 0 * 6 * 512 + 3 * 512);
  // depth 1
  pass_hi<<<512, 256, 0, stream>>>(w2, w2 + SDIM, PS, w2, w2 + SDIM, PS,
                                   gmats + 1 * 6 * 512);
  pass_lo<<<2048, 256, 0, stream>>>(w2, w2 + SDIM, PS, w1, w1 + SDIM, PS,
                                    nullptr, 0, gmats + 1 * 6 * 512 + 3 * 512);
  // depth 2 (final: interleaved output with ring CNOT folded in)
  pass_hi<<<512, 256, 0, stream>>>(w1, w1 + SDIM, PS, w1, w1 + SDIM, PS,
                                   gmats + 2 * 6 * 512);
  pass_lo<<<2048, 256, 0, stream>>>(w1, w1 + SDIM, PS, nullptr, nullptr, 0,
                                    out, 1, gmats + 2 * 6 * 512 + 3 * 512);
}